// TropicalBasicBlock_53429393162414
// MI455X (gfx1250) — compile-verified
//
#include <hip/hip_runtime.h>
#include <stdint.h>

// Tropical basic block, MI455X/gfx1250.
//   out = min( max_plus(min_plus(x,W1), W2)+b2 , max_plus(min_plus(x,Wm), Ws)+bs )
// VALU-bound (~805M f32 semiring ops vs ~4.5MB HBM traffic). WMMA is
// inapplicable (tropical semiring has no sum-of-products mapping), so we lean
// on the other CDNA5 paths: wave32 VALU (VOPD dual-issue), SMEM scalar
// broadcast of x, LDS broadcast of the intermediate tiles, and TDM
// (tensor_load_to_lds) async staging of the layer-2 weights, double-buffered
// behind compute and synchronized with s_wait_tensorcnt.

#define BATCH   1024
#define IN_DIM  512
#define OUT_DIM 256
#define ROWS    8              // batch rows per block
#define KC      32             // layer-2 K-chunk staged per TDM op
#define NCHUNK  (OUT_DIM / KC) // 8

typedef uint32_t v4u __attribute__((ext_vector_type(4)));
typedef int      v8i __attribute__((ext_vector_type(8)));
typedef int      v4i __attribute__((ext_vector_type(4)));

// Dynamic-LDS layout (byte offsets; no static LDS, so dynamic base == 0).
#define LDS_H1   0                                  // 8 x 256 f32  =  8KB
#define LDS_HM   (ROWS * OUT_DIM * 4)               //  8192
#define LDS_W2   (2 * ROWS * OUT_DIM * 4)           // 16384, 2 x (32x256) f32
#define LDS_WS   (LDS_W2 + 2 * KC * OUT_DIM * 4)    // 81920
#define LDS_SIZE (LDS_WS + 2 * KC * OUT_DIM * 4)    // 147456 B (2 blocks/WGP)

// Issue one TDM 2D tile load: `rows` x OUT_DIM f32, row stride OUT_DIM,
// from global `gptr` into LDS byte offset `lds_off`. D# packed per CDNA5 ISA
// ch.8 (group0: count/lds/global/type, group1: data_size + dims/strides).
// Verified against emitted asm: tensor_load_to_lds s[g0], s[g1].
__device__ __forceinline__ void tdm_load_tile(uint32_t lds_off, const float* gptr,
                                              uint32_t rows) {
  uint64_t ga = (uint64_t)(uintptr_t)gptr;
  const uint32_t td0 = OUT_DIM, tile0 = OUT_DIM, s0 = OUT_DIM;
  const uint32_t td1 = rows,    tile1 = rows;

  v4u g0;
  g0[0] = 1u;                                             // count=1, user D#
  g0[1] = lds_off;                                        // lds_addr [63:32]
  g0[2] = (uint32_t)ga;                                   // global_addr[31:0]
  g0[3] = (uint32_t)((ga >> 32) & 0x1FFFFFFu) | (2u << 30); // addr[56:32], type=2

  v8i g1;
  g1[0] = (int)(2u << 16);                                // data_size=4B, wg_mask=0
  g1[1] = (int)((td0 & 0xFFFFu) << 16);                   // tensor_dim0[15:0] @bit48
  g1[2] = (int)((td0 >> 16) | ((td1 & 0xFFFFu) << 16));   // td0[31:16] | td1[15:0]
  g1[3] = (int)((td1 >> 16) | (tile0 << 16));             // td1[31:16] | tile_dim0
  g1[4] = (int)tile1;                                     // tile_dim1 (tile_dim2=0)
  g1[5] = (int)s0;                                        // tensor_dim0_stride[31:0]
  g1[6] = 0;                                              // stride0[47:32], stride1 lo
  g1[7] = 0;                                              // stride1 hi (2D: unused)

  v4i z4 = {0, 0, 0, 0};                                  // groups 2/3: 2D tensor
  v8i z8 = {0, 0, 0, 0, 0, 0, 0, 0};                      // extra group (unused)
  __builtin_amdgcn_tensor_load_to_lds(g0, g1, z4, z4, z8, /*cpol=*/0);
}

__global__ __launch_bounds__(256) void tropical_block_kernel(
    const float* __restrict__ x,  const float* __restrict__ W1,
    const float* __restrict__ W2, const float* __restrict__ b2,
    const float* __restrict__ Wm, const float* __restrict__ Ws,
    const float* __restrict__ bs, float* __restrict__ out) {
  extern __shared__ char smem[];
  float* lds = (float*)smem;

  const int col  = threadIdx.x;          // 0..255: output column (lane-varying)
  const int wave = threadIdx.x >> 5;
  const int row0 = blockIdx.x * ROWS;

  // Kick off async TDM staging of the first layer-2 weight chunks; the whole
  // 512-deep layer-1 loop hides this DMA.
  if (wave == 0) {
    tdm_load_tile(LDS_W2, W2, KC);
    tdm_load_tile(LDS_WS, Ws, KC);
  }

  const float INF = __builtin_inff();
  float acc1[ROWS], accm[ROWS];
#pragma unroll
  for (int r = 0; r < ROWS; ++r) { acc1[r] = INF; accm[r] = INF; }

  // ---- Layer 1: fused (min,+) for main + shortcut paths.
  // x[(row0+r)*IN+k] is wave-uniform -> SMEM s_load broadcast (free VGPR port);
  // W1/Wm loads are fully coalesced b32 per lane.
#pragma unroll 4
  for (int k = 0; k < IN_DIM; ++k) {
    float w1v = W1[k * OUT_DIM + col];
    float wmv = Wm[k * OUT_DIM + col];
#pragma unroll
    for (int r = 0; r < ROWS; ++r) {
      float xv = x[(row0 + r) * IN_DIM + k];   // uniform scalar load
      acc1[r] = fminf(acc1[r], xv + w1v);
      accm[r] = fminf(accm[r], xv + wmv);
    }
  }

  // Stash intermediate tiles for the layer-2 K-reduction (read as broadcasts).
#pragma unroll
  for (int r = 0; r < ROWS; ++r) {
    lds[LDS_H1 / 4 + r * OUT_DIM + col] = acc1[r];
    lds[LDS_HM / 4 + r * OUT_DIM + col] = accm[r];
  }
  if (wave == 0) __builtin_amdgcn_s_wait_tensorcnt(0);  // chunk 0 landed
  __syncthreads();                                      // h tiles + W chunk 0 visible

  float outv[ROWS], scv[ROWS];
#pragma unroll
  for (int r = 0; r < ROWS; ++r) { outv[r] = -INF; scv[r] = -INF; }

  // ---- Layer 2: fused (max,+) for both paths, TDM double-buffered weights.
#pragma unroll 1
  for (int c = 0; c < NCHUNK; ++c) {
    const int buf = c & 1;
    if (wave == 0 && (c + 1) < NCHUNK) {   // prefetch next chunk into spare buffer
      tdm_load_tile(LDS_W2 + (buf ^ 1) * KC * OUT_DIM * 4, W2 + (c + 1) * KC * OUT_DIM, KC);
      tdm_load_tile(LDS_WS + (buf ^ 1) * KC * OUT_DIM * 4, Ws + (c + 1) * KC * OUT_DIM, KC);
    }
    const float* w2b = lds + LDS_W2 / 4 + buf * KC * OUT_DIM;
    const float* wsb = lds + LDS_WS / 4 + buf * KC * OUT_DIM;
    const float* h1b = lds + LDS_H1 / 4 + c * KC;   // + r*OUT_DIM + k
    const float* hmb = lds + LDS_HM / 4 + c * KC;

    // unroll 2 (not full): each body already exposes ~64 independent VALU ops;
    // full unroll only inflates live ranges and causes scratch spills.
#pragma unroll 2
    for (int k4 = 0; k4 < KC; k4 += 4) {
      // Per-lane weight reads from LDS (stride OUT_DIM between k's).
      float w2v[4], wsv[4];
#pragma unroll
      for (int u = 0; u < 4; ++u) {
        w2v[u] = w2b[(k4 + u) * OUT_DIM + col];
        wsv[u] = wsb[(k4 + u) * OUT_DIM + col];
      }
#pragma unroll
      for (int r = 0; r < ROWS; ++r) {
        // Uniform-address float4 LDS broadcasts of the h tiles.
        float4 h1q = *(const float4*)(h1b + r * OUT_DIM + k4);
        float4 hmq = *(const float4*)(hmb + r * OUT_DIM + k4);
        outv[r] = fmaxf(outv[r], h1q.x + w2v[0]);
        outv[r] = fmaxf(outv[r], h1q.y + w2v[1]);
        outv[r] = fmaxf(outv[r], h1q.z + w2v[2]);
        outv[r] = fmaxf(outv[r], h1q.w + w2v[3]);
        scv[r]  = fmaxf(scv[r],  hmq.x + wsv[0]);
        scv[r]  = fmaxf(scv[r],  hmq.y + wsv[1]);
        scv[r]  = fmaxf(scv[r],  hmq.z + wsv[2]);
        scv[r]  = fmaxf(scv[r],  hmq.w + wsv[3]);
      }
    }
    if (wave == 0) __builtin_amdgcn_s_wait_tensorcnt(0); // next chunk landed
    __syncthreads();                                     // release buffers/chunk
  }

  // ---- Epilogue: bias add + residual min-combine, coalesced stores.
  const float b2v = b2[col];
  const float bsv = bs[col];
#pragma unroll
  for (int r = 0; r < ROWS; ++r) {
    out[(row0 + r) * OUT_DIM + col] = fminf(outv[r] + b2v, scv[r] + bsv);
  }
}

extern "C" void kernel_launch(void* const* d_in, const int* in_sizes, int n_in,
                              void* d_out, int out_size, void* d_ws, size_t ws_size,
                              hipStream_t stream) {
  (void)in_sizes; (void)n_in; (void)out_size; (void)d_ws; (void)ws_size;
  const float* x  = (const float*)d_in[0];
  const float* W1 = (const float*)d_in[1];
  const float* W2 = (const float*)d_in[2];
  const float* b2 = (const float*)d_in[3];
  const float* Wm = (const float*)d_in[4];
  const float* Ws = (const float*)d_in[5];
  const float* bs = (const float*)d_in[6];
  float* out = (float*)d_out;

  dim3 grid(BATCH / ROWS);   // 128 blocks
  dim3 block(OUT_DIM);       // 256 threads = 8 waves (wave32)
  hipLaunchKernelGGL(tropical_block_kernel, grid, block, LDS_SIZE, stream,
                     x, W1, W2, b2, Wm, Ws, bs, out);
}